// SimpleT5Model_59614146069093
// MI455X (gfx1250) — compile-verified
//
#include <hip/hip_runtime.h>

#define BN 8
#define TN 512
#define VN 22
#define CN 512

typedef __attribute__((ext_vector_type(2))) float v2f;
typedef __attribute__((ext_vector_type(8))) float v8f;

// ---------------------------------------------------------------------------
// Kernel 1: per-(b,t) mean over V joints + squared norms of the averaged rows
// One block per (b,t); 256 threads cover C=512 channels (2 each).
// ---------------------------------------------------------------------------
__global__ void avg_norm_kernel(const float* __restrict__ query,
                                const float* __restrict__ key,
                                float* __restrict__ avg_q,
                                float* __restrict__ avg_k,
                                float* __restrict__ qnorm,
                                float* __restrict__ knorm) {
  const int bt  = blockIdx.x;          // b*T + t
  const int tid = threadIdx.x;         // 0..255
  const float inv = 1.0f / (float)VN;
  const float* qb = query + (size_t)bt * VN * CN;
  const float* kb = key   + (size_t)bt * VN * CN;
  float qs = 0.f, ks = 0.f;
  for (int c = tid; c < CN; c += 256) {
    float aq = 0.f, ak = 0.f;
    for (int v = 0; v < VN; ++v) {
      aq += qb[v * CN + c];
      ak += kb[v * CN + c];
    }
    aq *= inv; ak *= inv;
    avg_q[(size_t)bt * CN + c] = aq;
    avg_k[(size_t)bt * CN + c] = ak;
    qs += aq * aq;
    ks += ak * ak;
  }
  __shared__ float r1[256], r2[256];
  r1[tid] = qs; r2[tid] = ks;
  __syncthreads();
  for (int off = 128; off > 0; off >>= 1) {
    if (tid < off) { r1[tid] += r1[tid + off]; r2[tid] += r2[tid + off]; }
    __syncthreads();
  }
  if (tid == 0) { qnorm[bt] = r1[0]; knorm[bt] = r2[0]; }
}

// ---------------------------------------------------------------------------
// Kernel 2: zero the cross-correlation accumulator (must happen every call)
// ---------------------------------------------------------------------------
__global__ void zero_kernel(float* __restrict__ p, int n) {
  int i = blockIdx.x * blockDim.x + threadIdx.x;
  if (i < n) p[i] = 0.f;
}

// ---------------------------------------------------------------------------
// Kernel 3: cross[b,s] = sum_{t<L} avg_q[b,t,:] . avg_k[b,t+s,:]
// = diagonal sums of the gram matrix M = avg_q * avg_k^T.
// One block per (b, 16-row strip of t). 8 waves each own 4 column tiles;
// each tile is a 16x16x512 fp32 GEMM built from V_WMMA_F32_16X16X4_F32.
// fp32 A 16x4 layout: lane = M (mod 16); VGPR j holds K = j + 2*(lane>=16).
// => per K-step each lane loads a contiguous float2 at offset k + 2*(lane>>4).
// Diagonals accumulate into LDS (ds_add_f32), then into global cross.
// ---------------------------------------------------------------------------
__global__ void cross_kernel(const float* __restrict__ avg_q,
                             const float* __restrict__ avg_k,
                             const int* __restrict__ seq_len,
                             float* __restrict__ cross) {
  __shared__ float ldsCross[TN];
  const int b    = blockIdx.x >> 5;    // T/16 == 32 row strips per batch
  const int ti   = blockIdx.x & 31;
  const int t0   = ti * 16;
  const int tid  = threadIdx.x;        // 0..255 (8 waves of 32)
  const int wave = tid >> 5;
  const int lane = tid & 31;
  for (int i = tid; i < TN; i += 256) ldsCross[i] = 0.f;
  __syncthreads();

  const int L = seq_len[b];
  const float* Q = avg_q + (size_t)b * TN * CN;
  const float* K = avg_k + (size_t)b * TN * CN;

  const int n     = lane & 15;          // A: M row / B: N col / D: N col
  const int khalf = (lane >> 4) << 1;   // K sub-offset: 0 (lanes 0-15) or 2
  const int mbase = (lane >> 4) << 3;   // D rows: 0..7 or 8..15
  const float* arow = Q + (size_t)(t0 + n) * CN + khalf;

  for (int si = wave; si < TN / 16; si += 8) {   // uniform per wave: EXEC all-1
    const int u0 = si * 16;
    const float* brow = K + (size_t)(u0 + n) * CN + khalf;
    v8f acc = {0.f, 0.f, 0.f, 0.f, 0.f, 0.f, 0.f, 0.f};
    for (int k = 0; k < CN; k += 4) {
      v2f a  = *(const v2f*)(arow + k);
      v2f bv = *(const v2f*)(brow + k);
      // D = A(16x4) * B(4x16) + C ; fp32
      acc = __builtin_amdgcn_wmma_f32_16x16x4_f32(
          false, a, false, bv, (short)0, acc, false, false);
    }
    // D[m,n] contributes to cross[s = u - t] when u >= t and t < L
    for (int i = 0; i < 8; ++i) {
      const int t = t0 + mbase + i;
      const int u = u0 + n;
      if (t < L && u >= t) atomicAdd(&ldsCross[u - t], acc[i]);
    }
  }
  __syncthreads();
  for (int s = tid; s < TN; s += 256)
    atomicAdd(&cross[b * TN + s], ldsCross[s]);
}

// ---------------------------------------------------------------------------
// Kernel 4: per-batch dist[s] = qsum + ksum[s] - 2*cross[s] over valid s,
// argmin with first-index tie-break -> start[b]. One block per batch.
// ---------------------------------------------------------------------------
__global__ void argmin_kernel(const float* __restrict__ qnorm,
                              const float* __restrict__ knorm,
                              const float* __restrict__ cross,
                              const int* __restrict__ seq_len,
                              int* __restrict__ startBuf) {
  const int b   = blockIdx.x;
  const int tid = threadIdx.x;       // 0..255
  const int L   = seq_len[b];
  __shared__ float csk[TN + 1];
  __shared__ float rv[256];
  __shared__ int   ri[256];

  // masked qsum
  float qs = 0.f;
  for (int t = tid; t < TN; t += 256)
    if (t < L) qs += qnorm[b * TN + t];
  rv[tid] = qs;
  __syncthreads();
  for (int off = 128; off > 0; off >>= 1) {
    if (tid < off) rv[tid] += rv[tid + off];
    __syncthreads();
  }
  const float qsum = rv[0];

  // serial cumsum of knorm (512 adds; negligible)
  if (tid == 0) {
    float a = 0.f;
    csk[0] = 0.f;
    for (int i = 0; i < TN; ++i) { a += knorm[b * TN + i]; csk[i + 1] = a; }
  }
  __syncthreads();

  float best = __builtin_inff();
  int bestIdx = TN;
  for (int s = tid; s < TN; s += 256) {
    float d;
    if (s <= TN - L) {
      int end = s + L; if (end > TN) end = TN;
      d = qsum + (csk[end] - csk[s]) - 2.0f * cross[b * TN + s];
    } else {
      d = __builtin_inff();
    }
    if (d < best || (d == best && s < bestIdx)) { best = d; bestIdx = s; }
  }
  rv[tid] = best; ri[tid] = bestIdx;
  __syncthreads();
  for (int off = 128; off > 0; off >>= 1) {
    if (tid < off) {
      if (rv[tid + off] < rv[tid] ||
          (rv[tid + off] == rv[tid] && ri[tid + off] < ri[tid])) {
        rv[tid] = rv[tid + off];
        ri[tid] = ri[tid + off];
      }
    }
    __syncthreads();
  }
  if (tid == 0) startBuf[b] = ri[0];
}

// ---------------------------------------------------------------------------
// Kernel 5: fused gather + interpolate + transpose + concat.
// One block per (b,v,t); 128 threads, float4 per thread over C=512.
// out[b,v,t, 0:C]   = query[b,t,v,:]
// out[b,v,t, C:2C]  = interp of sub = query - key[start+i]  (0 where unwritten)
// ---------------------------------------------------------------------------
__global__ void output_kernel(const float* __restrict__ query,
                              const float* __restrict__ key,
                              const int* __restrict__ seq_len,
                              const int* __restrict__ startBuf,
                              float* __restrict__ out) {
  int idx = blockIdx.x;
  const int t = idx % TN; idx /= TN;
  const int v = idx % VN; idx /= VN;
  const int b = idx;
  const int c0 = threadIdx.x << 2;   // 128 threads * 4 floats = 512

  const size_t qoff = (((size_t)b * TN + t) * VN + v) * CN + c0;
  float4 q4 = *(const float4*)(query + qoff);
  const size_t ooff = (((size_t)b * VN + v) * TN + t) * (size_t)(2 * CN) + c0;
  *(float4*)(out + ooff) = q4;

  const int L = seq_len[b];
  const int step = TN / L;
  float4 a4 = make_float4(0.f, 0.f, 0.f, 0.f);
  if (t < (L - 1) * step) {
    const int start = startBuf[b];
    int i0 = t / step;
    int i1 = i0 + 1; if (i1 > TN - 1) i1 = TN - 1;
    const float ratio = (float)(t % step) / (float)step;
    int g0 = start + i0; if (g0 > TN - 1) g0 = TN - 1;
    int g1 = start + i1; if (g1 > TN - 1) g1 = TN - 1;
    const size_t vc = (size_t)v * CN + c0;
    float4 q0 = *(const float4*)(query + ((size_t)b * TN + i0) * VN * CN + vc);
    float4 q1 = *(const float4*)(query + ((size_t)b * TN + i1) * VN * CN + vc);
    float4 k0 = *(const float4*)(key   + ((size_t)b * TN + g0) * VN * CN + vc);
    float4 k1 = *(const float4*)(key   + ((size_t)b * TN + g1) * VN * CN + vc);
    const float s0x = q0.x - k0.x, s0y = q0.y - k0.y,
                s0z = q0.z - k0.z, s0w = q0.w - k0.w;
    const float s1x = q1.x - k1.x, s1y = q1.y - k1.y,
                s1z = q1.z - k1.z, s1w = q1.w - k1.w;
    a4.x = s0x + ratio * (s1x - s0x);
    a4.y = s0y + ratio * (s1y - s0y);
    a4.z = s0z + ratio * (s1z - s0z);
    a4.w = s0w + ratio * (s1w - s0w);
  }
  *(float4*)(out + ooff + CN) = a4;
}

// ---------------------------------------------------------------------------
extern "C" void kernel_launch(void* const* d_in, const int* in_sizes, int n_in,
                              void* d_out, int out_size, void* d_ws, size_t ws_size,
                              hipStream_t stream) {
  const float* query   = (const float*)d_in[0];
  const float* key     = (const float*)d_in[1];
  const int*   seq_len = (const int*)d_in[2];
  float* out = (float*)d_out;

  // workspace layout (floats): avg_q | avg_k | qnorm | knorm | cross | start
  float* ws = (float*)d_ws;
  const size_t BTC = (size_t)BN * TN * CN;   // 2,097,152
  const size_t BT  = (size_t)BN * TN;        // 4,096
  float* avg_q = ws;
  float* avg_k = avg_q + BTC;
  float* qnorm = avg_k + BTC;
  float* knorm = qnorm + BT;
  float* cross = knorm + BT;
  int*   startBuf = (int*)(cross + BT);

  avg_norm_kernel<<<BN * TN, 256, 0, stream>>>(query, key, avg_q, avg_k,
                                               qnorm, knorm);
  zero_kernel<<<(int)((BT + 255) / 256), 256, 0, stream>>>(cross, (int)BT);
  cross_kernel<<<BN * (TN / 16), 256, 0, stream>>>(avg_q, avg_k, seq_len, cross);
  argmin_kernel<<<BN, 256, 0, stream>>>(qnorm, knorm, cross, seq_len, startBuf);
  output_kernel<<<BN * VN * TN, 128, 0, stream>>>(query, key, seq_len,
                                                  startBuf, out);
}